// WaveConvLayer_19121194402128
// MI455X (gfx1250) — compile-verified
//
#include <hip/hip_runtime.h>
#include <hip/hip_bf16.h>

typedef __attribute__((ext_vector_type(2))) float v2f;
typedef __attribute__((ext_vector_type(8))) float v8f;

// ---------------------------------------------------------------------------
// Kernel 1: real 3x3 conv, pad=1.  u:(16,64,32,32)  g:(64,64,3,3) -> out same.
// Tiny FLOP share; scalar FMA, cache-resident.
// blockDim = (32,8); grid = (16*64, 4)
// ---------------------------------------------------------------------------
__global__ void conv3x3_kernel(const float* __restrict__ u,
                               const float* __restrict__ g,
                               float* __restrict__ out) {
    const int w  = threadIdx.x;            // 0..31
    const int oh = blockIdx.y * 8 + threadIdx.y;  // 0..31
    const int nco = blockIdx.x;            // n*64 + co
    const int n  = nco >> 6;
    const int co = nco & 63;

    float acc = 0.f;
    for (int ci = 0; ci < 64; ++ci) {
        const float* up = u + (((size_t)n * 64 + ci) * 32) * 32;
        const float* gp = g + ((size_t)co * 64 + ci) * 9;
#pragma unroll
        for (int dy = 0; dy < 3; ++dy) {
            const int ih = oh + dy - 1;
            if ((unsigned)ih < 32u) {
#pragma unroll
                for (int dx = 0; dx < 3; ++dx) {
                    const int iw = w + dx - 1;
                    if ((unsigned)iw < 32u)
                        acc = fmaf(gp[dy * 3 + dx], up[ih * 32 + iw], acc);
                }
            }
        }
    }
    out[(((size_t)n * 64 + co) * 32 + oh) * 32 + w] = acc;
}

// ---------------------------------------------------------------------------
// Kernel 2: complex 1x1 "gain" conv as fp32 WMMA GEMM.
//   u: (16, 64, 6, H, H, 2)   g: (6, 2, 64, 64)   out: (16, 64, 6, H, H, 2)
// Per scale s: V[co,p] = sum_ci G[s,co,ci] * U[ci,p]   (complex)
// One wave computes one 16(co) x 16(spatial) complex tile, K=64 in steps of 4.
// 4 x V_WMMA_F32_16X16X4_F32 per K-step (complex multiply).
// grid: waves = 6 * 4 * 16 * H * (H/16), blockDim = 256 (8 waves).
// ---------------------------------------------------------------------------
__global__ void cgain_wmma_kernel(const float* __restrict__ u,
                                  const float* __restrict__ g,
                                  float* __restrict__ out,
                                  int H) {
    const int WT   = H >> 4;
    const int wave = (int)((blockIdx.x * blockDim.x + threadIdx.x) >> 5);
    const int lane = (int)(threadIdx.x & 31);

    int t = wave;
    const int wt = t % WT; t /= WT;
    const int h  = t % H;  t /= H;
    const int n  = t % 16; t /= 16;
    const int m  = t % 4;  t /= 4;   // co tile (16 channels)
    const int s  = t;                // scale 0..5

    const int half = lane >> 4;      // 0: K pair {0,1}; 1: K pair {2,3}
    const int lm   = lane & 15;
    const int p    = wt * 16 + lm;   // spatial column (w coordinate)
    const int co   = m * 16 + lm;    // A-fragment row

    // float2 (complex) units. idx(n,c,s,h,p) = ((n*64+c)*6+s)*H*H + h*H + p
    const size_t cstride = (size_t)6 * H * H;                       // per-channel
    const size_t base    = ((size_t)n * 64 * 6 + s) * H * H + (size_t)h * H + p;
    const float2* ub = (const float2*)u + base;
    float2*       ob = (float2*)out + base;

    const float* grp = g + (((size_t)s * 2 + 0) * 64 + co) * 64;    // Re weights row
    const float* gip = g + (((size_t)s * 2 + 1) * 64 + co) * 64;    // Im weights row

    v8f accr = {0.f, 0.f, 0.f, 0.f, 0.f, 0.f, 0.f, 0.f};
    v8f acci = {0.f, 0.f, 0.f, 0.f, 0.f, 0.f, 0.f, 0.f};

    for (int k0 = 0; k0 < 64; k0 += 4) {
        const int kc = k0 + 2 * half;           // this lane's K pair base
        // A fragments (weights), contiguous ci pair -> float2 load
        const float2 gr2 = *(const float2*)(grp + kc);
        const float2 gi2 = *(const float2*)(gip + kc);
        v2f Ar  = { gr2.x,  gr2.y};
        v2f Ai  = { gi2.x,  gi2.y};
        v2f nAi = {-gi2.x, -gi2.y};
        // B fragments (activations): two channels, complex pair each
        const float2 z0 = ub[(size_t)kc * cstride];
        const float2 z1 = ub[(size_t)(kc + 1) * cstride];
        v2f Br = {z0.x, z1.x};
        v2f Bi = {z0.y, z1.y};

        accr = __builtin_amdgcn_wmma_f32_16x16x4_f32(false, Ar,  false, Br,
                                                     (short)0, accr, false, false);
        accr = __builtin_amdgcn_wmma_f32_16x16x4_f32(false, nAi, false, Bi,
                                                     (short)0, accr, false, false);
        acci = __builtin_amdgcn_wmma_f32_16x16x4_f32(false, Ai,  false, Br,
                                                     (short)0, acci, false, false);
        acci = __builtin_amdgcn_wmma_f32_16x16x4_f32(false, Ar,  false, Bi,
                                                     (short)0, acci, false, false);
    }

    // D layout: VGPR j holds (M = j + 8*half, N = lm); store {re, im} pairs.
#pragma unroll
    for (int j = 0; j < 8; ++j) {
        const int co_out = m * 16 + j + 8 * half;
        float2 v; v.x = accr[j]; v.y = acci[j];
        ob[(size_t)co_out * cstride] = v;
    }
}

// ---------------------------------------------------------------------------
// Launch
// ---------------------------------------------------------------------------
extern "C" void kernel_launch(void* const* d_in, const int* in_sizes, int n_in,
                              void* d_out, int out_size, void* d_ws, size_t ws_size,
                              hipStream_t stream) {
    const float* u_lp = (const float*)d_in[0];  // (16,64,32,32)
    const float* u0   = (const float*)d_in[1];  // (16,64,6,64,64,2)
    const float* u1   = (const float*)d_in[2];  // (16,64,6,32,32,2)
    const float* g_lp = (const float*)d_in[3];  // (64,64,3,3)
    const float* g0   = (const float*)d_in[4];  // (6,2,64,64,1,1)
    const float* g1   = (const float*)d_in[5];  // (6,2,64,64,1,1)

    float* v_lp = (float*)d_out;
    float* v0   = v_lp + (size_t)16 * 64 * 32 * 32;
    float* v1   = v0   + (size_t)16 * 64 * 6 * 64 * 64 * 2;

    // v_lp: 3x3 conv
    conv3x3_kernel<<<dim3(16 * 64, 4), dim3(32, 8), 0, stream>>>(u_lp, g_lp, v_lp);

    // v0: H=64 -> waves = 6*4*16*64*4 = 98304 -> 12288 blocks of 256
    cgain_wmma_kernel<<<12288, 256, 0, stream>>>(u0, g0, v0, 64);

    // v1: H=32 -> waves = 6*4*16*32*2 = 24576 -> 3072 blocks of 256
    cgain_wmma_kernel<<<3072, 256, 0, stream>>>(u1, g1, v1, 32);
}